// LocalGlobalScaledDotProductAttention_72026601554186
// MI455X (gfx1250) — compile-verified
//
#include <hip/hip_runtime.h>

// ---------------- problem constants (match reference) ----------------
#define DM     512      // DMID
#define SS     4096     // S
#define BB     2        // B
#define NH     8        // HEADS
#define HDIM   64       // HD
#define GG     64       // G (global tokens)
#define WHALF  256      // W (half window)
#define NCOL   (64 + 2*WHALF + 16)   // 592 = sel(64) + window(528)

#define KB     32       // k-slab staged per TDM transfer (floats)
#define KPAD   36       // LDS row stride for k-slabs (TDM pad: +4 dw / 32 dw)
#define NKB    (DM / KB)

typedef __attribute__((ext_vector_type(2))) float        v2f;
typedef __attribute__((ext_vector_type(8))) float        v8f;
typedef __attribute__((ext_vector_type(4))) unsigned int v4u;
typedef __attribute__((ext_vector_type(4))) int          v4i;
typedef __attribute__((ext_vector_type(8))) int          v8i;

#ifndef __has_builtin
#define __has_builtin(x) 0
#endif
#if __has_builtin(__builtin_amdgcn_tensor_load_to_lds) && \
    __has_builtin(__builtin_amdgcn_s_wait_tensorcnt)
#define USE_TDM 1
#else
#define USE_TDM 0
#endif

// D = A(16x4 f32) * B(4x16 f32) + C(16x16 f32), wave32.
// A: lane L -> row M=L%16, VGPR j -> K=2*(L/16)+j  (contiguous K pair / lane)
// B: lane L -> col N=L%16, VGPR j -> K=2*(L/16)+j
// C/D: VGPR r -> row r+8*(L/16), col L%16
static __device__ __forceinline__ v8f wmma4(v2f a, v2f b, v8f c) {
  return __builtin_amdgcn_wmma_f32_16x16x4_f32(false, a, false, b, (short)0, c,
                                               false, false);
}
static __device__ __forceinline__ v8f zero8() {
  v8f z = {0.f, 0.f, 0.f, 0.f, 0.f, 0.f, 0.f, 0.f};
  return z;
}

#if USE_TDM
static __device__ __forceinline__ unsigned lds_off(const void* p) {
  // generic pointer to LDS: low 32 bits are the LDS byte offset
  return (unsigned)(unsigned long long)p;
}
// 2D tile load global->LDS via Tensor Data Mover (ISA 08, D# per §8.3/8.4).
// tile_w elements per row, tile_h rows, row stride in elements, f32 data.
// pad_icode/pad_acode: LDS padding (interval 2^(i+1) dwords, amount a+1 dwords)
static __device__ __forceinline__ void tdm_load_2d(
    unsigned ldsoff, const void* gp, unsigned tile_w, unsigned tile_h,
    unsigned stride_elems, unsigned pad_icode, unsigned pad_acode, bool pad_en)
{
  const unsigned long long ga = (unsigned long long)gp;
  v4u g0;
  g0.x = 1u;                                   // count=1, user mode
  g0.y = ldsoff;                               // lds_addr (bytes)
  g0.z = (unsigned)ga;                         // global_addr[31:0]
  g0.w = (unsigned)((ga >> 32) & 0x01ffffffu)  // global_addr[56:32]
         | 0x80000000u;                        // type=2 ("image")
  unsigned w0 = (2u << 16);                    // data_size = 4 bytes
  if (pad_en) w0 |= (1u << 20) | (pad_icode << 22) | (pad_acode << 25);
  const unsigned td = 0x40000000u;             // huge tensor dims: no OOB clip
  v8i g1;
  g1[0] = (int)w0;
  g1[1] = (int)((td & 0xffffu) << 16);               // tensor_dim0 lo16
  g1[2] = (int)((td >> 16) | ((td & 0xffffu) << 16)); // dim0 hi16 | dim1 lo16
  g1[3] = (int)((td >> 16) | (tile_w << 16));        // dim1 hi16 | tile_dim0
  g1[4] = (int)(tile_h & 0xffffu);                   // tile_dim1 (tile_dim2=0)
  g1[5] = (int)stride_elems;                         // tensor_dim0_stride lo32
  g1[6] = 0;
  g1[7] = 0;
  v4i gz = {0, 0, 0, 0};
#if __clang_major__ >= 23
  v8i gz8 = {0, 0, 0, 0, 0, 0, 0, 0};
  __builtin_amdgcn_tensor_load_to_lds(g0, g1, gz, gz, gz8, 0);
#else
  __builtin_amdgcn_tensor_load_to_lds(g0, g1, gz, gz, 0);
#endif
}
#endif // USE_TDM

// ---------------------------------------------------------------------
// Kernel 1: Y[z,h,row,hd] = (X[z,row,:] . W[n,:] + bias[n]) * scale
// 4-wave workgroup computes a 64x64 tile; X/W k-slabs (64x32 f32) staged in
// LDS by the TDM, double-buffered on TENSORcnt; 8 f32 WMMA k-steps/slab,
// A operand reused across 4 N-tiles (4 accumulators per wave).
// ---------------------------------------------------------------------
__global__ __launch_bounds__(128) void proj_wmma(
    const float* __restrict__ X, long xbs, int ldx,
    const float* __restrict__ Wt, const float* __restrict__ bias,
    float scale, float* __restrict__ Y, long ybs, int Mout)
{
  __shared__ float lX[2][64][KPAD];
  __shared__ float lW[2][64][KPAD];

  const int n0   = blockIdx.x * 64;
  const int m0   = blockIdx.y * 64;
  const int z    = blockIdx.z;
  const int tid  = threadIdx.x;
  const int wave = tid >> 5;
  const int li   = tid & 15;
  const int hf   = (tid >> 4) & 1;

  const float* Xb = X + (long)z * xbs + (long)m0 * ldx;
  const float* Wb = Wt + (long)n0 * DM;

  v8f acc0 = zero8(), acc1 = zero8(), acc2 = zero8(), acc3 = zero8();

  auto stage = [&](int bu, int kblk) {
#if USE_TDM
    if (wave == 0)
      tdm_load_2d(lds_off(&lX[bu][0][0]), Xb + kblk, KB, 64, (unsigned)ldx,
                  4, 3, true);                 // pad 4 dw every 32 dw
    else if (wave == 1)
      tdm_load_2d(lds_off(&lW[bu][0][0]), Wb + kblk, KB, 64, DM, 4, 3, true);
#else
    for (int i = tid; i < 64 * (KB / 2); i += 128) {
      const int r = i / (KB / 2), c = (i % (KB / 2)) * 2;
      *(v2f*)&lX[bu][r][c] = *(const v2f*)(Xb + (long)r * ldx + kblk + c);
      *(v2f*)&lW[bu][r][c] = *(const v2f*)(Wb + (long)r * DM + kblk + c);
    }
#endif
  };

  stage(0, 0);
  for (int t = 0; t < NKB; ++t) {
    if (t + 1 < NKB) {
      stage((t + 1) & 1, (t + 1) * KB);
#if USE_TDM
      __builtin_amdgcn_s_wait_tensorcnt((short)1);  // prev slab done
#endif
    } else {
#if USE_TDM
      __builtin_amdgcn_s_wait_tensorcnt((short)0);
#endif
    }
    __syncthreads();
    const int bu = t & 1;
#pragma unroll
    for (int kb = 0; kb < KB; kb += 4) {
      const int c = kb + 2 * hf;
      v2f a  = *(const v2f*)&lX[bu][wave * 16 + li][c];
      v2f b0 = *(const v2f*)&lW[bu][ 0 + li][c];
      v2f b1 = *(const v2f*)&lW[bu][16 + li][c];
      v2f b2 = *(const v2f*)&lW[bu][32 + li][c];
      v2f b3 = *(const v2f*)&lW[bu][48 + li][c];
      acc0 = wmma4(a, b0, acc0);
      acc1 = wmma4(a, b1, acc1);
      acc2 = wmma4(a, b2, acc2);
      acc3 = wmma4(a, b3, acc3);
    }
    __syncthreads();
  }

  const int hh = n0 / HDIM;                  // n0 is a multiple of HDIM
  float* Yb = Y + (long)z * ybs + (long)hh * Mout * HDIM;
  const float bv0 = bias[n0 +  0 + li];
  const float bv1 = bias[n0 + 16 + li];
  const float bv2 = bias[n0 + 32 + li];
  const float bv3 = bias[n0 + 48 + li];
#pragma unroll
  for (int r = 0; r < 8; ++r) {
    const int row = m0 + wave * 16 + r + 8 * hf;
    float* yr = Yb + (long)row * HDIM;
    yr[ 0 + li] = (acc0[r] + bv0) * scale;
    yr[16 + li] = (acc1[r] + bv1) * scale;
    yr[32 + li] = (acc2[r] + bv2) * scale;
    yr[48 + li] = (acc3[r] + bv3) * scale;
  }
}

// ---------------------------------------------------------------------
// Kernel 2: local banded attention + 64 global keys, for all S queries.
// One wave per (b, h, 16-query tile). Q tile TDM-staged into LDS (read 37x).
// Scores staged in LDS (16 x 592 f32):
//   cols [0,64)   : sel scores vs global keys 0..63
//   cols [64,592) : window keys s0-W .. s0+15+W; out-of-band -> -1e9,
//                   in-band keys with attn_mask!=0 -> +(-10000)
// Row softmax over 592 cols, then out(16x64) = P(16x592) x Vcat(592x64).
// ---------------------------------------------------------------------
__global__ __launch_bounds__(32) void local_attn(
    const float* __restrict__ qh, const float* __restrict__ kh,
    const float* __restrict__ vh, const int* __restrict__ am,
    float* __restrict__ out)
{
  __shared__ float sc[16][NCOL];
  __shared__ float lQ[16][68];   // 64 + 4 pad -> bank-conflict-free rows
  const int s0   = blockIdx.x * 16;
  const int h    = blockIdx.y;
  const int b    = blockIdx.z;
  const int lane = threadIdx.x;
  const int li   = lane & 15;
  const int hf   = lane >> 4;

  const long   base = (long)(b * NH + h) * SS * HDIM;
  const float* Q = qh + base;
  const float* K = kh + base;
  const float* V = vh + base;

#if USE_TDM
  tdm_load_2d(lds_off(&lQ[0][0]), Q + (long)s0 * HDIM, 64, 16, HDIM,
              5, 3, true);                    // pad 4 dw every 64 dw
  __builtin_amdgcn_s_wait_tensorcnt((short)0);
#else
  for (int i = lane; i < 16 * 32; i += 32) {
    const int r = i >> 5, c = (i & 31) * 2;
    *(v2f*)&lQ[r][c] = *(const v2f*)(Q + (long)(s0 + r) * HDIM + c);
  }
#endif
  __syncthreads();

  // ---- sel scores: keys 0..63 -> cols 0..63
#pragma unroll
  for (int kt = 0; kt < 4; ++kt) {
    v8f acc = zero8();
    const float* Kr = K + (long)(kt * 16 + li) * HDIM + 2 * hf;
    for (int kb = 0; kb < HDIM; kb += 4) {
      v2f a  = *(const v2f*)&lQ[li][kb + 2 * hf];
      v2f bm = *(const v2f*)(Kr + kb);
      acc = wmma4(a, bm, acc);
    }
#pragma unroll
    for (int r = 0; r < 8; ++r) sc[r + 8 * hf][kt * 16 + li] = acc[r];
  }

  // ---- band scores: 33 key tiles covering s0-W .. s0+15+W
  for (int kt = 0; kt < 33; ++kt) {
    const int  cc   = kt * 16 + li;       // 0..527
    const int  key  = s0 - WHALF + cc;
    const bool inr  = (key >= 0) && (key < SS);
    const int  keyc = inr ? key : 0;
    v8f acc = zero8();
    const float* Kr = K + (long)keyc * HDIM + 2 * hf;
    for (int kb = 0; kb < HDIM; kb += 4) {
      v2f a  = *(const v2f*)&lQ[li][kb + 2 * hf];
      v2f bm = *(const v2f*)(Kr + kb);
      acc = wmma4(a, bm, acc);
    }
    const float fmv = (am[b * SS + keyc] != 0) ? -10000.f : 0.f;
#pragma unroll
    for (int r = 0; r < 8; ++r) {
      const int  rr    = r + 8 * hf;
      const int  rel   = cc - rr;         // = key - (s0+rr) + W
      const bool valid = inr && (rel >= 0) && (rel <= 2 * WHALF);
      sc[rr][64 + cc] = valid ? (acc[r] + fmv) : -1.0e9f;
    }
  }
  __syncthreads();

  // ---- row softmax (2 lanes per row, cross-half reduce via shfl_xor 16)
  {
    const int row = li;
    float mx = -3.0e38f;
    for (int c = hf; c < NCOL; c += 2) mx = fmaxf(mx, sc[row][c]);
    mx = fmaxf(mx, __shfl_xor(mx, 16));
    float sum = 0.f;
    for (int c = hf; c < NCOL; c += 2) {
      float e = __expf(sc[row][c] - mx);
      sc[row][c] = e;
      sum += e;
    }
    sum += __shfl_xor(sum, 16);
    const float inv = (am[b * SS + s0 + row] < 0) ? 0.f : (1.f / sum);
    for (int c = hf; c < NCOL; c += 2) sc[row][c] *= inv;
  }
  __syncthreads();

  // ---- out(16x64) = P(16x592) x Vcat(592x64); Vcat row rv:
  //      rv < 64  -> vh[key=rv];  rv >= 64 -> vh[key=s0-W+(rv-64)] (clamped)
#pragma unroll
  for (int nt = 0; nt < 4; ++nt) {
    v8f acc = zero8();
    for (int kb = 0; kb < NCOL; kb += 4) {   // 148 WMMA k-steps
      const int k0 = kb + 2 * hf;
      v2f a = *(const v2f*)&sc[li][k0];
      int key0 = (k0 < 64) ? k0 : (s0 - WHALF + (k0 - 64));
      int key1 = (k0 + 1 < 64) ? (k0 + 1) : (s0 - WHALF + (k0 + 1 - 64));
      key0 = min(max(key0, 0), SS - 1);
      key1 = min(max(key1, 0), SS - 1);
      v2f bm;
      bm.x = V[(long)key0 * HDIM + nt * 16 + li];
      bm.y = V[(long)key1 * HDIM + nt * 16 + li];
      acc = wmma4(a, bm, acc);
    }
#pragma unroll
    for (int r = 0; r < 8; ++r) {
      const int rr = r + 8 * hf;
      out[((long)b * SS + s0 + rr) * DM + h * HDIM + nt * 16 + li] = acc[r];
    }
  }
}

// ---------------------------------------------------------------------
// Kernel 3: dense global attention for the first G rows (overwrites them).
// One wave per (b, h, 16 global queries), flash-style online softmax with
// WMMA C-operand accumulation; Q tile TDM-staged (read 256x).
// ---------------------------------------------------------------------
__global__ __launch_bounds__(32) void global_attn(
    const float* __restrict__ qg, const float* __restrict__ kg,
    const float* __restrict__ vg, const int* __restrict__ am,
    float* __restrict__ out)
{
  __shared__ float st[16][16];
  __shared__ float pt[16][16];
  __shared__ float lQ[16][68];
  __shared__ float rowm[16], rowl[16], rowf[16];
  const int g0   = blockIdx.x * 16;
  const int h    = blockIdx.y;
  const int b    = blockIdx.z;
  const int lane = threadIdx.x;
  const int li   = lane & 15;
  const int hf   = lane >> 4;

  const float* QG = qg + (long)(b * NH + h) * GG * HDIM;
  const float* KG = kg + (long)(b * NH + h) * SS * HDIM;
  const float* VG = vg + (long)(b * NH + h) * SS * HDIM;

#if USE_TDM
  tdm_load_2d(lds_off(&lQ[0][0]), QG + (long)g0 * HDIM, 64, 16, HDIM,
              5, 3, true);
  __builtin_amdgcn_s_wait_tensorcnt((short)0);
#else
  for (int i = lane; i < 16 * 32; i += 32) {
    const int r = i >> 5, c = (i & 31) * 2;
    *(v2f*)&lQ[r][c] = *(const v2f*)(QG + (long)(g0 + r) * HDIM + c);
  }
#endif

  v8f acc0 = zero8(), acc1 = zero8(), acc2 = zero8(), acc3 = zero8();
  if (lane < 16) { rowm[lane] = -3.0e38f; rowl[lane] = 0.f; }
  __syncthreads();

  for (int t = 0; t < SS / 16; ++t) {
    const int kt0 = t * 16;

    v8f s = zero8();
    const float* Kr = KG + (long)(kt0 + li) * HDIM + 2 * hf;
    for (int kb = 0; kb < HDIM; kb += 4) {
      v2f a  = *(const v2f*)&lQ[li][kb + 2 * hf];
      v2f bm = *(const v2f*)(Kr + kb);
      s = wmma4(a, bm, s);
    }
    const float fmv = (am[b * SS + kt0 + li] < 0) ? -10000.f : 0.f;
#pragma unroll
    for (int r = 0; r < 8; ++r) st[r + 8 * hf][li] = s[r] + fmv;
    __syncthreads();

    {   // online softmax update (2 lanes per row)
      const int row = li;
      const int c0  = hf * 8;
      float tmx = -3.0e38f;
#pragma unroll
      for (int c = 0; c < 8; ++c) tmx = fmaxf(tmx, st[row][c0 + c]);
      tmx = fmaxf(tmx, __shfl_xor(tmx, 16));
      const float m_old = rowm[row];
      const float m_new = fmaxf(m_old, tmx);
      const float f     = __expf(m_old - m_new);
      float ps = 0.f;
#pragma unroll
      for (int c = 0; c < 8; ++c) {
        float e = __expf(st[row][c0 + c] - m_new);
        pt[row][c0 + c] = e;
        ps += e;
      }
      ps += __shfl_xor(ps, 16);
      if (hf == 0) {
        rowm[row] = m_new;
        rowl[row] = rowl[row] * f + ps;
        rowf[row] = f;
      }
    }
    __syncthreads();

#pragma unroll
    for (int r = 0; r < 8; ++r) {
      const float fr = rowf[r + 8 * hf];
      acc0[r] *= fr; acc1[r] *= fr; acc2[r] *= fr; acc3[r] *= fr;
    }
#pragma unroll
    for (int kb = 0; kb < 16; kb += 4) {
      const int k0 = kb + 2 * hf;
      v2f a = *(const v2f*)&pt[li][k0];
      const float* vr0 = VG + (long)(kt0 + k0) * HDIM + li;
      const float* vr1 = VG + (long)(kt0 + k0 + 1) * HDIM + li;
      v2f b0; b0.x = vr0[0];  b0.y = vr1[0];  acc0 = wmma4(a, b0, acc0);
      v2f b1; b1.x = vr0[16]; b1.y = vr1[16]; acc1 = wmma4(a, b1, acc1);
      v2f b2; b2.x = vr0[32]; b2.y = vr1[32]; acc2 = wmma4(a, b2, acc2);
      v2f b3; b3.x = vr0[48]; b3.y = vr1[48]; acc3 = wmma4(a, b3, acc3);
    }
    __syncthreads();
  }

#pragma unroll
  for (int r = 0; r < 8; ++r) {
    const int   rr  = r + 8 * hf;
    const float inv = 1.f / rowl[rr];
    const long  o   = ((long)b * SS + g0 + rr) * DM + h * HDIM;
    out[o +  0 + li] = acc0[r] * inv;
    out[o + 16 + li] = acc1[r] * inv;
    out[o + 32 + li] = acc2[r] * inv;
    out[o + 48 + li] = acc3[r] * inv;
  }
}

// ---------------------------------------------------------------------
extern "C" void kernel_launch(void* const* d_in, const int* in_sizes, int n_in,
                              void* d_out, int out_size, void* d_ws,
                              size_t ws_size, hipStream_t stream)
{
  (void)in_sizes; (void)n_in; (void)out_size; (void)ws_size;

  const float* hs  = (const float*)d_in[0];   // (B,S,DM)
  const float* q   = (const float*)d_in[1];   // (S,B,DM)
  const float* k   = (const float*)d_in[2];   // (S,B,DM)
  const float* v   = (const float*)d_in[3];   // (S,B,DM)
  const int*   am  = (const int*)d_in[4];     // (B,S) int32
  const float* Wq  = (const float*)d_in[5];
  const float* bq  = (const float*)d_in[6];
  const float* Wk  = (const float*)d_in[7];
  const float* bk  = (const float*)d_in[8];
  const float* Wv  = (const float*)d_in[9];
  const float* bv  = (const float*)d_in[10];
  const float* Wqg = (const float*)d_in[11];
  const float* bqg = (const float*)d_in[12];
  const float* Wkg = (const float*)d_in[13];
  const float* bkg = (const float*)d_in[14];
  const float* Wvg = (const float*)d_in[15];
  const float* bvg = (const float*)d_in[16];
  float* out = (float*)d_out;

  // workspace: five (B,H,S,HD) buffers + one (B,H,G,HD) buffer = ~84.1 MB
  const long bigN = (long)BB * NH * SS * HDIM;   // 4,194,304 floats
  float* ws = (float*)d_ws;
  float* qh = ws;
  float* kh = qh + bigN;
  float* vh = kh + bigN;
  float* kg = vh + bigN;
  float* vg = kg + bigN;
  float* qg = vg + bigN;                          // B*NH*GG*HDIM floats

  const float scale = 0.125f;                     // 1/sqrt(64)
  const long  ybsS = (long)NH * SS * HDIM;
  const long  ybsG = (long)NH * GG * HDIM;

  dim3 blk(128, 1, 1);
  dim3 gp(DM / 64, SS / 64, BB);
  // q,k,v are (S,B,DM): batch stride DM, row stride B*DM
  proj_wmma<<<gp, blk, 0, stream>>>(q, (long)DM, BB * DM, Wq, bq, scale, qh, ybsS, SS);
  proj_wmma<<<gp, blk, 0, stream>>>(k, (long)DM, BB * DM, Wk, bk, 1.f,  kh, ybsS, SS);
  proj_wmma<<<gp, blk, 0, stream>>>(v, (long)DM, BB * DM, Wv, bv, 1.f,  vh, ybsS, SS);
  // hidden_states is (B,S,DM): batch stride S*DM, row stride DM
  proj_wmma<<<gp, blk, 0, stream>>>(hs, (long)SS * DM, DM, Wkg, bkg, 1.f, kg, ybsS, SS);
  proj_wmma<<<gp, blk, 0, stream>>>(hs, (long)SS * DM, DM, Wvg, bvg, 1.f, vg, ybsS, SS);
  dim3 gq(DM / 64, GG / 64, BB);
  proj_wmma<<<gq, blk, 0, stream>>>(hs, (long)SS * DM, DM, Wqg, bqg, scale, qg, ybsG, GG);

  dim3 ba(32, 1, 1);
  dim3 gb(SS / 16, NH, BB);   // 256 x 8 x 2 waves
  local_attn<<<gb, ba, 0, stream>>>(qh, kh, vh, am, out);

  dim3 gc(GG / 16, NH, BB);   // 4 x 8 x 2 waves, overwrites rows < G
  global_attn<<<gc, ba, 0, stream>>>(qg, kg, vg, am, out);
}